// LDPCNetwork_28991029248395
// MI455X (gfx1250) — compile-verified
//
#include <hip/hip_runtime.h>
#include <cstdint>
#include <cstddef>

#define N_VN    24576
#define D_V     4
#define D_C     8
#define E_EDGES (N_VN * D_V)     // 98304
#define M_CN    (E_EDGES / D_C)  // 12288
#define N_ITERS 10
#define BSZ     64
#define CLIPV   20.0f
#define EPS_F   1e-12f

__device__ __forceinline__ float clipf(float x, float lo, float hi) {
    return fminf(fmaxf(x, lo), hi);
}

// tanh(0.5*x) using the gfx1250 hardware tanh if the builtin is exposed,
// otherwise via hw v_exp_f32 + v_rcp_f32.
__device__ __forceinline__ float tanh_half(float x) {
#if __has_builtin(__builtin_amdgcn_tanhf)
    return __builtin_amdgcn_tanhf(0.5f * x);
#elif __has_builtin(__builtin_amdgcn_tanh_f32)
    return __builtin_amdgcn_tanh_f32(0.5f * x);
#else
    float ex = __expf(x);                       // v_exp_f32
    return __fdividef(ex - 1.0f, ex + 1.0f);    // v_rcp_f32 + v_mul
#endif
}

// ---------------------------------------------------------------------------
// init: transpose llr_in (64 x N) -> llr_t (N x 64), zero sum_llr/c2v/vn_count
// ---------------------------------------------------------------------------
__global__ __launch_bounds__(256) void ldpc_init(
    const float* __restrict__ llr_in, float* __restrict__ llr_t,
    float* __restrict__ sum_llr, float* __restrict__ c2v,
    int* __restrict__ vn_count)
{
    int i = blockIdx.x * 256 + threadIdx.x;
    if (i < E_EDGES * BSZ) c2v[i] = 0.0f;
    if (i < N_VN * BSZ) {
        sum_llr[i] = 0.0f;
        int n = i >> 6;
        int b = i & 63;
        llr_t[i] = llr_in[b * N_VN + n];
    }
    if (i < N_VN) vn_count[i] = 0;
}

// ---------------------------------------------------------------------------
// build inverse map vn -> its 4 edges (atomic append), then sort for a
// deterministic summation order.
// ---------------------------------------------------------------------------
__global__ __launch_bounds__(256) void ldpc_build(
    const int* __restrict__ edge_to_vn, int* __restrict__ vn_count,
    int* __restrict__ vn_edges)
{
    int e = blockIdx.x * 256 + threadIdx.x;
    if (e < E_EDGES) {
        int v = edge_to_vn[e];
        int p = atomicAdd(&vn_count[v], 1);
        vn_edges[v * 4 + p] = e;
    }
}

__global__ __launch_bounds__(256) void ldpc_sort(int* __restrict__ vn_edges)
{
    int n = blockIdx.x * 256 + threadIdx.x;
    if (n >= N_VN) return;
    int a0 = vn_edges[n * 4 + 0], a1 = vn_edges[n * 4 + 1];
    int a2 = vn_edges[n * 4 + 2], a3 = vn_edges[n * 4 + 3];
    int t;
    if (a0 > a1) { t = a0; a0 = a1; a1 = t; }
    if (a2 > a3) { t = a2; a2 = a3; a3 = t; }
    if (a0 > a2) { t = a0; a0 = a2; a2 = t; }
    if (a1 > a3) { t = a1; a1 = a3; a3 = t; }
    if (a1 > a2) { t = a1; a1 = a2; a2 = t; }
    vn_edges[n * 4 + 0] = a0; vn_edges[n * 4 + 1] = a1;
    vn_edges[n * 4 + 2] = a2; vn_edges[n * 4 + 3] = a3;
}

// ---------------------------------------------------------------------------
// CN kernel: block = 4 check nodes x 64 batch lanes (8 wave32 waves).
// The block's old-c2v tile (4 CN * 8 edges * 64 batch * 4B = 8 KB) is a single
// contiguous region -> stage it into LDS with the CDNA5 async DMA path
// (global_load_async_to_lds_b128 + s_wait_asynccnt), then compute:
//   v2c  = clip(ch_w*llr + sum_llr - c2v_old)
//   t    = tanh(v2c/2)                      (hw v_tanh_f32 / v_exp_f32)
//   excl = prefix*suffix product of the other 7 t's in the CN
//   c2v' = clip(cn_w * clip(log((1+p)/(1-p+eps))))
// ---------------------------------------------------------------------------
#define CN_PER_BLK 4

__global__ __launch_bounds__(256) void ldpc_cn(
    const float* __restrict__ llr_t, const float* __restrict__ sum_llr,
    float* __restrict__ c2v, const int* __restrict__ edge_to_vn,
    const float* __restrict__ cn_weight, const float* __restrict__ ch_weight,
    int it)
{
    __shared__ float lds_c2v[CN_PER_BLK * D_C * BSZ];   // 8 KB

    const int b   = threadIdx.x;                 // batch lane 0..63
    const int j   = threadIdx.y;                 // cn within block 0..3
    const int tid = threadIdx.y * 64 + threadIdx.x;
    const int cn0 = blockIdx.x * CN_PER_BLK;

    // ---- async stage old c2v tile: 512 x b128 ops, 2 per thread ----
    const float* gbase = c2v + (size_t)cn0 * (D_C * BSZ);
    unsigned lbase = (unsigned)(size_t)(void*)&lds_c2v[0]; // LDS byte offset
    #pragma unroll
    for (int k = 0; k < 2; ++k) {
        unsigned slot = (unsigned)tid + (unsigned)k * 256u;
        unsigned goff = slot * 16u;
        unsigned loff = lbase + slot * 16u;
        asm volatile(
            "global_load_async_to_lds_b128 %0, %1, %2 offset:0"
            :
            : "v"(loff), "v"(goff), "s"(gbase)
            : "memory");
    }
    asm volatile("s_wait_asynccnt 0" ::: "memory");
    __syncthreads();

    const float chw = ch_weight[it];
    const float cnw = cn_weight[it];
    const int   eb  = (cn0 + j) * D_C;

    float t[D_C];
    #pragma unroll
    for (int k = 0; k < D_C; ++k) {
        int e = eb + k;
        // edge_to_vn[e] is uniform across the wave (y fixed within a wave):
        // force it onto the scalar path so the gathers use saddr form.
        int v = __builtin_amdgcn_readfirstlane(edge_to_vn[e]);
        float old = lds_c2v[(j * D_C + k) * BSZ + b];
        float x = fmaf(chw, llr_t[(size_t)v * BSZ + b],
                       sum_llr[(size_t)v * BSZ + b]) - old;
        x = clipf(x, -CLIPV, CLIPV);
        float th = tanh_half(x);
        th = (th == 0.0f) ? EPS_F : th;
        t[k] = th;
    }

    // exclusive products over the 8 edges of this CN (prefix * suffix)
    float pre[D_C];
    pre[0] = 1.0f;
    #pragma unroll
    for (int k = 1; k < D_C; ++k) pre[k] = pre[k - 1] * t[k - 1];

    float suf = 1.0f;
    #pragma unroll
    for (int k = D_C - 1; k >= 0; --k) {
        float p = pre[k] * suf;
        suf *= t[k];
        p = clipf(p, -0.999999f, 0.999999f);
        float o = __logf(__fdividef(1.0f + p, 1.0f - p + EPS_F)); // v_log_f32
        o = clipf(o, -CLIPV, CLIPV);
        o = clipf(o * cnw, -CLIPV, CLIPV);
        c2v[(size_t)(eb + k) * BSZ + b] = o;
    }
}

// ---------------------------------------------------------------------------
// VN kernel: sum_llr[n][b] = sum of c2v over the VN's 4 (sorted) edges.
// ---------------------------------------------------------------------------
__global__ __launch_bounds__(256) void ldpc_vn(
    const float* __restrict__ c2v, const int* __restrict__ vn_edges,
    float* __restrict__ sum_llr)
{
    int b = threadIdx.x;
    int n = blockIdx.x * 4 + threadIdx.y;
    const int4 e4 = *reinterpret_cast<const int4*>(vn_edges + (size_t)n * 4);
    // indices are uniform per wave -> scalarize for saddr-form gathers
    int e0 = __builtin_amdgcn_readfirstlane(e4.x);
    int e1 = __builtin_amdgcn_readfirstlane(e4.y);
    int e2 = __builtin_amdgcn_readfirstlane(e4.z);
    int e3 = __builtin_amdgcn_readfirstlane(e4.w);
    float s = c2v[(size_t)e0 * BSZ + b];
    s += c2v[(size_t)e1 * BSZ + b];
    s += c2v[(size_t)e2 * BSZ + b];
    s += c2v[(size_t)e3 * BSZ + b];
    sum_llr[(size_t)n * BSZ + b] = s;
}

// ---------------------------------------------------------------------------
// final: dec_llr[b][n] = llr_in[b][n] + sum_llr[n][b]   (transpose back)
// ---------------------------------------------------------------------------
__global__ __launch_bounds__(256) void ldpc_final(
    const float* __restrict__ llr_in, const float* __restrict__ sum_llr,
    float* __restrict__ out)
{
    int i = blockIdx.x * 256 + threadIdx.x;
    if (i < N_VN * BSZ) {
        int b = i / N_VN;
        int n = i - b * N_VN;
        out[i] = llr_in[i] + sum_llr[(size_t)n * BSZ + b];
    }
}

// ---------------------------------------------------------------------------
extern "C" void kernel_launch(void* const* d_in, const int* in_sizes, int n_in,
                              void* d_out, int out_size, void* d_ws, size_t ws_size,
                              hipStream_t stream)
{
    const float* llr_in     = (const float*)d_in[0];
    const float* cn_weight  = (const float*)d_in[1];
    const float* ch_weight  = (const float*)d_in[2];
    const int*   edge_to_vn = (const int*)d_in[3];
    // d_in[4] (edge_to_ext_edge) is structurally deterministic: edge e's
    // extrinsic set is the other 7 edges of check node e/8 -> not needed.
    float* out = (float*)d_out;

    constexpr size_t SZ_NB   = (size_t)N_VN * BSZ * sizeof(float);    // 6291456
    constexpr size_t SZ_EB   = (size_t)E_EDGES * BSZ * sizeof(float); // 25165824
    char* ws = (char*)d_ws;
    float* llr_t    = (float*)(ws);
    float* sum_llr  = (float*)(ws + SZ_NB);
    float* c2v      = (float*)(ws + 2 * SZ_NB);
    int*   vn_edges = (int*)  (ws + 2 * SZ_NB + SZ_EB);
    int*   vn_count = (int*)  (ws + 2 * SZ_NB + SZ_EB + (size_t)N_VN * 4 * sizeof(int));

    ldpc_init<<<(E_EDGES * BSZ + 255) / 256, 256, 0, stream>>>(
        llr_in, llr_t, sum_llr, c2v, vn_count);
    ldpc_build<<<(E_EDGES + 255) / 256, 256, 0, stream>>>(
        edge_to_vn, vn_count, vn_edges);
    ldpc_sort<<<(N_VN + 255) / 256, 256, 0, stream>>>(vn_edges);

    for (int it = 0; it < N_ITERS; ++it) {
        ldpc_cn<<<M_CN / CN_PER_BLK, dim3(64, CN_PER_BLK), 0, stream>>>(
            llr_t, sum_llr, c2v, edge_to_vn, cn_weight, ch_weight, it);
        ldpc_vn<<<N_VN / 4, dim3(64, 4), 0, stream>>>(c2v, vn_edges, sum_llr);
    }

    ldpc_final<<<(N_VN * BSZ + 255) / 256, 256, 0, stream>>>(
        llr_in, sum_llr, out);
}